// Net_14422500180428
// MI455X (gfx1250) — compile-verified
//
#include <hip/hip_runtime.h>
#include <cmath>

typedef __attribute__((ext_vector_type(2))) float v2f;
typedef __attribute__((ext_vector_type(8))) float v8f;

#define D_FEAT 512
#define HIDDEN 16   // == N_CLASSES

// ---------------------------------------------------------------- zero fill
__global__ void k_zero(float* __restrict__ p, long long n) {
  long long i = (long long)blockIdx.x * blockDim.x + threadIdx.x;
  long long stride = (long long)gridDim.x * blockDim.x;
  for (; i < n; i += stride) p[i] = 0.0f;
}

// ---------------------------------------------------------------- layer-1 GEMM
// h[N,16] = x[N,512] @ W[512,16], one wave32 per 16-row tile, f32 WMMA K=4.
__global__ void k_gemm1(const float* __restrict__ x, const float* __restrict__ W,
                        float* __restrict__ h, int ntiles) {
  const int wave = blockIdx.x * (blockDim.x >> 5) + (threadIdx.x >> 5);
  if (wave >= ntiles) return;                    // wave-uniform: EXEC stays all-1s
  const int lane = threadIdx.x & 31;
  const int half = lane >> 4;                    // 0: K=k0..k0+1, 1: K=k0+2..k0+3
  const int l16  = lane & 15;

  const float* xrow = x + ((size_t)wave * 16 + l16) * D_FEAT + 2 * half; // A: M=l16
  const float* wptr = W + (2 * half) * HIDDEN + l16;                     // B: N=l16

  v8f c = {};
#pragma unroll 4
  for (int k0 = 0; k0 < D_FEAT; k0 += 4) {
    v2f a = *(const v2f*)(xrow + k0);            // x[M][k0+2h], x[M][k0+2h+1]
    v2f b;
    b.x = wptr[k0 * HIDDEN];                     // W[k0+2h  ][N]
    b.y = wptr[(k0 + 1) * HIDDEN];               // W[k0+2h+1][N]
    c = __builtin_amdgcn_wmma_f32_16x16x4_f32(false, a, false, b,
                                              (short)0, c, false, false);
  }
  // C/D: VGPR v -> row (v + 8*half), col l16
  float* out = h + ((size_t)wave * 16 + 8 * half) * HIDDEN + l16;
#pragma unroll
  for (int v = 0; v < 8; ++v) out[v * HIDDEN] = c[v];
}

// ---------------------------------------------------------------- layer-2 GEMM
// h2[N,16] = relu(acc + b1) @ W2[16,16]
__global__ void k_gemm2(const float* __restrict__ acc, const float* __restrict__ bias,
                        const float* __restrict__ W, float* __restrict__ h, int ntiles) {
  const int wave = blockIdx.x * (blockDim.x >> 5) + (threadIdx.x >> 5);
  if (wave >= ntiles) return;
  const int lane = threadIdx.x & 31;
  const int half = lane >> 4;
  const int l16  = lane & 15;

  const float* arow = acc + ((size_t)wave * 16 + l16) * HIDDEN;

  v8f c = {};
#pragma unroll
  for (int k0 = 0; k0 < HIDDEN; k0 += 4) {
    const int ka = k0 + 2 * half;
    v2f a, b;
    a.x = fmaxf(arow[ka]     + bias[ka],     0.0f);   // relu(out1 + b1)
    a.y = fmaxf(arow[ka + 1] + bias[ka + 1], 0.0f);
    b.x = W[ka * HIDDEN + l16];
    b.y = W[(ka + 1) * HIDDEN + l16];
    c = __builtin_amdgcn_wmma_f32_16x16x4_f32(false, a, false, b,
                                              (short)0, c, false, false);
  }
  float* out = h + ((size_t)wave * 16 + 8 * half) * HIDDEN + l16;
#pragma unroll
  for (int v = 0; v < 8; ++v) out[v * HIDDEN] = c[v];
}

// ---------------------------------------------------------------- edge scatter
// out[dst[e]][f] += w[e] * h[src[e]][f]; 16 consecutive lanes share an edge
__global__ void k_scatter(const float* __restrict__ h, const int* __restrict__ src,
                          const int* __restrict__ dst, const float* __restrict__ w,
                          float* __restrict__ out, long long nedges) {
  long long i = (long long)blockIdx.x * blockDim.x + threadIdx.x;
  const long long stride = (long long)gridDim.x * blockDim.x;
  const long long total = nedges * HIDDEN;
  for (; i < total; i += stride) {
    const long long e = i >> 4;
    const int f = (int)(i & 15);
    const int s = src[e];
    const int d = dst[e];
    const float v = w[e] * h[(size_t)s * HIDDEN + f];
    atomicAdd(out + (size_t)d * HIDDEN + f, v);
  }
}

// ---------------------------------------------------------------- bias + log_softmax
__global__ void k_final(const float* __restrict__ acc, const float* __restrict__ bias,
                        float* __restrict__ out, int n) {
  const int i = blockIdx.x * blockDim.x + threadIdx.x;
  if (i >= n) return;
  float v[HIDDEN];
  float m = -INFINITY;
#pragma unroll
  for (int f = 0; f < HIDDEN; ++f) {
    v[f] = acc[(size_t)i * HIDDEN + f] + bias[f];
    m = fmaxf(m, v[f]);
  }
  float s = 0.0f;
#pragma unroll
  for (int f = 0; f < HIDDEN; ++f) s += expf(v[f] - m);
  const float lse = m + logf(s);
#pragma unroll
  for (int f = 0; f < HIDDEN; ++f) out[(size_t)i * HIDDEN + f] = v[f] - lse;
}

// ---------------------------------------------------------------- launcher
extern "C" void kernel_launch(void* const* d_in, const int* in_sizes, int n_in,
                              void* d_out, int out_size, void* d_ws, size_t ws_size,
                              hipStream_t stream) {
  const float* x   = (const float*)d_in[0];
  const int*   ei  = (const int*)d_in[1];   // [2,E] int32 (JAX x64 off)
  const float* ew  = (const float*)d_in[2];
  const float* W1  = (const float*)d_in[3];
  const float* b1  = (const float*)d_in[4];
  const float* W2  = (const float*)d_in[5];
  const float* b2  = (const float*)d_in[6];

  const int       N = in_sizes[0] / D_FEAT;      // 100000
  const long long E = in_sizes[2];               // 3200000
  const int* src = ei;
  const int* dst = ei + E;

  float* bufA = (float*)d_ws;                    // h1, later h2   (N*16 f32)
  float* bufB = bufA + (size_t)N * HIDDEN;       // out1, later out2

  const int ntiles = N / 16;                     // 6250 (exact)
  const dim3 blk(256);                           // 8 waves/block
  const int gemm_blocks = (ntiles + 7) / 8;
  const long long nelem = (long long)N * HIDDEN;

  // out1 = 0
  k_zero<<<512, blk, 0, stream>>>(bufB, nelem);
  // h1 = x @ W1
  k_gemm1<<<gemm_blocks, blk, 0, stream>>>(x, W1, bufA, ntiles);
  // out1[dst] += w * h1[src]
  k_scatter<<<32768, blk, 0, stream>>>(bufA, src, dst, ew, bufB, E);
  // h2 = relu(out1 + b1) @ W2
  k_gemm2<<<gemm_blocks, blk, 0, stream>>>(bufB, b1, W2, bufA, ntiles);
  // out2 = 0
  k_zero<<<512, blk, 0, stream>>>(bufB, nelem);
  // out2[dst] += w * h2[src]
  k_scatter<<<32768, blk, 0, stream>>>(bufA, src, dst, ew, bufB, E);
  // log_softmax(out2 + b2)
  k_final<<<(N + 255) / 256, blk, 0, stream>>>(bufB, b2, (float*)d_out, N);
}